// PagedAttention_75041668596285
// MI455X (gfx1250) — compile-verified
//
#include <hip/hip_runtime.h>

// ---- problem constants (match reference) ----
constexpr int NSEQ  = 32;    // sequences
constexpr int NH    = 32;    // query heads
constexpr int NKV   = 8;     // kv heads
constexpr int GRP   = NH / NKV;   // 4 query heads per kv head
constexpr int HD    = 128;   // head size
constexpr int BLK   = 16;    // tokens per KV page
constexpr int MAXB  = 256;   // max pages per sequence
constexpr int NWAVE = 8;     // waves per workgroup (wave32)
constexpr float SCALE = 0.08838834764831845f; // 128^-0.5

typedef __attribute__((ext_vector_type(2))) float v2f;
typedef __attribute__((ext_vector_type(8))) float v8f;

// D(16x16,f32) = A(16x4,f32) x B(4x16,f32) + C   -- full-precision CDNA5 WMMA
__device__ __forceinline__ v8f wmma4(v2f a, v2f b, v8f c) {
  return __builtin_amdgcn_wmma_f32_16x16x4_f32(
      /*neg_a=*/false, a, /*neg_b=*/false, b,
      /*c_mod=*/(short)0, c, /*reuse_a=*/false, /*reuse_b=*/false);
}

__global__ __launch_bounds__(256, 1)
void PagedAttention_75041668596285_kernel(const float* __restrict__ q,
                                          const float* __restrict__ kc,
                                          const float* __restrict__ vc,
                                          const int*   __restrict__ bt,
                                          const int*   __restrict__ cl,
                                          float*       __restrict__ out) {
  // per-wave partials for the cross-wave merge
  __shared__ float lds_o[NWAVE][HD][GRP];  // 16 KB
  __shared__ float lds_m[NWAVE][GRP];
  __shared__ float lds_l[NWAVE][GRP];

  const int sq   = blockIdx.x / NKV;
  const int kvh  = blockIdx.x % NKV;
  const int tid  = threadIdx.x;
  const int w    = tid >> 5;        // wave id 0..7
  const int lane = tid & 31;
  const int lh   = lane & 15;       // N-column (head) / M-row within half
  const int hi   = lane >> 4;       // 0 = lower half, 1 = upper half

  const int ctx     = cl[sq];
  const int nblocks = (ctx + BLK - 1) / BLK;

  // ---- Q^T resident in registers as WMMA B-operands (pre-scaled) ----
  // chunk c covers dims k0=4c: lane<16 -> (Q[h][k0],Q[h][k0+1]),
  //                            lane>=16 -> (Q[h][k0+2],Q[h][k0+3]), h = lh (valid h<GRP)
  v2f qb[32];
  {
    const bool hv = (lh < GRP);
    const float* qrow = q + (((size_t)sq * NH) + (size_t)kvh * GRP + (hv ? lh : 0)) * HD;
    #pragma unroll
    for (int c = 0; c < 32; ++c) {
      v2f t = *(const v2f*)(qrow + 4 * c + 2 * hi);
      qb[c].x = hv ? t.x * SCALE : 0.0f;
      qb[c].y = hv ? t.y * SCALE : 0.0f;
    }
  }

  // O^T accumulator: 8 tiles of (16 dims x 16 head-cols); head index on lanes.
  v8f acc[8] = {};
  float m = -1e30f;   // running max  (per head == per lane, replicated across halves)
  float l = 0.0f;     // running sum

  for (int blk = w; blk < nblocks; blk += NWAVE) {
    const int pb = bt[(size_t)sq * MAXB + blk];
    const float* kb = kc + (((size_t)pb * NKV) + kvh) * (size_t)(BLK * HD);
    const float* vb = vc + (((size_t)pb * NKV) + kvh) * (size_t)(BLK * HD);

    // prefetch next page for this wave (global_prefetch_b8)
    if (blk + NWAVE < nblocks) {
      const int pbn = bt[(size_t)sq * MAXB + blk + NWAVE];
      const float* kn = kc + (((size_t)pbn * NKV) + kvh) * (size_t)(BLK * HD);
      const float* vn = vc + (((size_t)pbn * NKV) + kvh) * (size_t)(BLK * HD);
      __builtin_prefetch(kn + lane * 64, 0, 0);
      __builtin_prefetch(kn + lane * 64 + 32, 0, 0);
      __builtin_prefetch(vn + lane * 64, 0, 0);
      __builtin_prefetch(vn + lane * 64 + 32, 0, 0);
    }

    // ---- S^T(16tok x 16heads) = K(16x128) x Q^T(128x16), 32 WMMAs ----
    v8f sc = {};
    const float* krow = kb + lh * HD + 2 * hi;   // A: lane row = token lh, dim pair per half
    #pragma unroll
    for (int c = 0; c < 32; ++c) {
      v2f a = *(const v2f*)(krow + 4 * c);
      sc = wmma4(a, qb[c], sc);
    }

    // ---- online softmax (token index = reg r + 8*hi; head on lane) ----
    float p[8];
    const int tb = blk * BLK + (hi ? 8 : 0);
    float bm = -1e30f;
    #pragma unroll
    for (int r = 0; r < 8; ++r) {
      float sv = (tb + r < ctx) ? sc[r] : -1e30f;
      p[r] = sv;
      bm = fmaxf(bm, sv);
    }
    bm = fmaxf(bm, __shfl_xor(bm, 16, 32));       // combine token halves
    const float mnew  = fmaxf(m, bm);
    const float alpha = __expf(m - mnew);
    float psum = 0.0f;
    #pragma unroll
    for (int r = 0; r < 8; ++r) { p[r] = __expf(p[r] - mnew); psum += p[r]; }
    psum += __shfl_xor(psum, 16, 32);
    l = l * alpha + psum;
    m = mnew;
    #pragma unroll
    for (int i = 0; i < 8; ++i)
      #pragma unroll
      for (int r = 0; r < 8; ++r) acc[i][r] *= alpha;   // alpha is per-lane (per-head)

    // ---- O^T += V^T x P^T : 4 token-quads x 8 dim-tiles = 32 WMMAs ----
    #pragma unroll
    for (int t0 = 0; t0 < BLK; t0 += 4) {
      // B = P^T chunk: lane<16 -> (P^T[t0][lh],P^T[t0+1][lh]); lane>=16 -> (P^T[t0+2],P^T[t0+3])
      // P^T[t][h] lives in reg p[t&7] of lane h + 16*(t>=8)
      const int sl = lh + ((t0 & 8) ? 16 : 0);
      const float x_lo = __shfl(p[(t0    ) & 7], sl, 32);
      const float y_lo = __shfl(p[(t0 + 1) & 7], sl, 32);
      const float x_hi = __shfl(p[(t0 + 2) & 7], sl, 32);
      const float y_hi = __shfl(p[(t0 + 3) & 7], sl, 32);
      v2f bP;
      bP.x = hi ? x_hi : x_lo;
      bP.y = hi ? y_hi : y_lo;
      const float* v0 = vb + (size_t)(t0 + 2 * hi) * HD + lh;  // A = V^T chunk
      #pragma unroll
      for (int mt = 0; mt < 8; ++mt) {
        v2f a;
        a.x = v0[mt * 16];        // V[t][d0+lh]
        a.y = v0[mt * 16 + HD];   // V[t+1][d0+lh]
        acc[mt] = wmma4(a, bP, acc[mt]);
      }
    }
  }

  // ---- spill per-wave partials (only valid head columns) ----
  if (lh < GRP) {
    #pragma unroll
    for (int mt = 0; mt < 8; ++mt)
      #pragma unroll
      for (int r = 0; r < 8; ++r)
        lds_o[w][16 * mt + r + (hi ? 8 : 0)][lh] = acc[mt][r];
  }
  if (lane < GRP) { lds_m[w][lane] = m; lds_l[w][lane] = l; }
  __syncthreads();

  // ---- log-sum-exp merge across 8 waves; 512 outputs over 256 threads ----
  for (int e = tid; e < GRP * HD; e += 256) {
    const int h = e >> 7;          // 0..3
    const int d = e & (HD - 1);
    float M = -1e30f;
    for (int ww = 0; ww < NWAVE; ++ww) M = fmaxf(M, lds_m[ww][h]);
    float num = 0.0f, den = 0.0f;
    for (int ww = 0; ww < NWAVE; ++ww) {
      const float wgt = __expf(lds_m[ww][h] - M);
      num += wgt * lds_o[ww][d][h];
      den += wgt * lds_l[ww][h];
    }
    out[(((size_t)sq * NH) + (size_t)kvh * GRP + h) * HD + d] = num / den;
  }
}

extern "C" void kernel_launch(void* const* d_in, const int* in_sizes, int n_in,
                              void* d_out, int out_size, void* d_ws, size_t ws_size,
                              hipStream_t stream) {
  const float* q  = (const float*)d_in[0];
  const float* kc = (const float*)d_in[1];
  const float* vc = (const float*)d_in[2];
  const int*   bt = (const int*)d_in[3];
  const int*   cl = (const int*)d_in[4];
  float* out = (float*)d_out;
  PagedAttention_75041668596285_kernel<<<dim3(NSEQ * NKV), dim3(256), 0, stream>>>(
      q, kc, vc, bt, cl, out);
}